// PhACE_19121194402014
// MI455X (gfx1250) — compile-verified
//
#include <hip/hip_runtime.h>
#include <hip/hip_bf16.h>
#include <math.h>

#define N_ATOMS   10000
#define N_EDGES   160000
#define N_SPECIES 8
#define N_STRUCT  8
#define CCH       16      // channels C
#define NMAX      4
#define NBASIS    8
#define LMAX      3
#define CUTOFF    5.0f
#define KTOT      256     // (LMAX+1) * NMAX * CCH

typedef __attribute__((ext_vector_type(2))) float v2f;
typedef __attribute__((ext_vector_type(8))) float v8f;

// Native non-returning fp32 atomic add (CDNA5 GLOBAL_ATOMIC_ADD_F32).
// Guarantees the L2-resident fast path; avoids any CAS-loop lowering.
__device__ __forceinline__ void atomic_add_f32(float* p, float v) {
    asm volatile("global_atomic_add_f32 %0, %1, off"
                 :
                 : "v"(p), "v"(v)
                 : "memory");
}

// ---------------------------------------------------------------------------
// Zero a float buffer, b128 stores (n must be a multiple of 4).
// ---------------------------------------------------------------------------
__global__ void zero4_kernel(float4* __restrict__ p, size_t n4) {
    size_t i = (size_t)blockIdx.x * blockDim.x + threadIdx.x;
    size_t stride = (size_t)gridDim.x * blockDim.x;
    float4 z = {0.f, 0.f, 0.f, 0.f};
    for (; i < n4; i += stride) p[i] = z;
}

__global__ void zero_kernel(float* __restrict__ p, size_t n) {
    size_t i = (size_t)blockIdx.x * blockDim.x + threadIdx.x;
    if (i < n) p[i] = 0.0f;
}

// ---------------------------------------------------------------------------
// Per-edge geometry: spherical harmonics (l<=3) x radial basis -> phi[E,64]
// phi index j = lm*4 + n  where lm enumerates (l,m) pairs 0..15, n = 0..3.
// ---------------------------------------------------------------------------
__global__ void edge_phi_kernel(const float* __restrict__ pos,
                                const float* __restrict__ W_rad,   // [4][8][4]
                                const int*   __restrict__ senders,
                                const int*   __restrict__ receivers,
                                float*       __restrict__ phi) {
    int e = blockIdx.x * blockDim.x + threadIdx.x;
    if (e >= N_EDGES) return;
    int s = senders[e], r = receivers[e];
    float dx = pos[r * 3 + 0] - pos[s * 3 + 0];
    float dy = pos[r * 3 + 1] - pos[s * 3 + 1];
    float dz = pos[r * 3 + 2] - pos[s * 3 + 2];
    float rn = sqrtf(dx * dx + dy * dy + dz * dz);
    float rr = fmaxf(rn, 1e-6f);
    float inv_r = 1.0f / rr;
    float x = dx * inv_r, y = dy * inv_r, z = dz * inv_r;
    float x2 = x * x, y2 = y * y, z2 = z * z;

    float sh[16];
    sh[0]  = 0.28209479f;
    sh[1]  = 0.48860251f * y;
    sh[2]  = 0.48860251f * z;
    sh[3]  = 0.48860251f * x;
    sh[4]  = 1.09254843f * x * y;
    sh[5]  = 1.09254843f * y * z;
    sh[6]  = 0.31539157f * (3.0f * z2 - 1.0f);
    sh[7]  = 1.09254843f * x * z;
    sh[8]  = 0.54627422f * (x2 - y2);
    sh[9]  = 0.59004359f * y * (3.0f * x2 - y2);
    sh[10] = 2.89061144f * x * y * z;
    sh[11] = 0.45704579f * y * (5.0f * z2 - 1.0f);
    sh[12] = 0.37317633f * z * (5.0f * z2 - 3.0f);
    sh[13] = 0.45704579f * x * (5.0f * z2 - 1.0f);
    sh[14] = 1.44530572f * z * (x2 - y2);
    sh[15] = 0.59004359f * x * (x2 - 3.0f * y2);

    // Bessel radial basis with cosine cutoff
    const float PI = 3.14159265358979f;
    float fc = 0.5f * (cosf(PI * fminf(rn / CUTOFF, 1.0f)) + 1.0f);
    float scale = sqrtf(2.0f / CUTOFF) * fc;
    float basis[NBASIS];
#pragma unroll
    for (int b = 0; b < NBASIS; ++b) {
        float nn = (float)(b + 1);
        basis[b] = scale * sinf(nn * PI * rr / CUTOFF) * inv_r;
    }
    float R[LMAX + 1][NMAX];
#pragma unroll
    for (int l = 0; l <= LMAX; ++l) {
#pragma unroll
        for (int n = 0; n < NMAX; ++n) {
            float acc = 0.0f;
#pragma unroll
            for (int b = 0; b < NBASIS; ++b)
                acc += basis[b] * W_rad[(l * NBASIS + b) * NMAX + n];
            R[l][n] = acc;
        }
    }

    // phi[e, lm*4+n] = sh[lm] * R[l(lm)][n]; contiguous float4 per lm
    float4* pe4 = reinterpret_cast<float4*>(phi + (size_t)e * 64);
    int lm = 0;
#pragma unroll
    for (int l = 0; l <= LMAX; ++l) {
        for (int m = 0; m < 2 * l + 1; ++m) {
            float sv = sh[l * l + m];
            float4 v = {sv * R[l][0], sv * R[l][1], sv * R[l][2], sv * R[l][3]};
            pe4[lm] = v;
            ++lm;
        }
    }
}

// ---------------------------------------------------------------------------
// cemb[a,c] = embed[species[a], c]
// ---------------------------------------------------------------------------
__global__ void cemb_kernel(const float* __restrict__ embed,
                            const int*   __restrict__ species,
                            float*       __restrict__ cemb) {
    int i = blockIdx.x * blockDim.x + threadIdx.x;
    if (i >= N_ATOMS * CCH) return;
    int a = i >> 4, c = i & 15;
    cemb[i] = embed[species[a] * CCH + c];
}

// ---------------------------------------------------------------------------
// Edge scatter: one wave32 per edge.
// A[r, j, c] += phi[e, j] * h[s, c]   (j = 0..63, c = 0..15)
// Lane layout: c = lane&15, half = lane>>4 handles j in [half*32, half*32+32).
// Per lane: 8x global_load_b128 of phi + 32x global_atomic_add_f32.
// A is 40 MB -> L2-resident; native fp32 atomics resolve in L2.
// ---------------------------------------------------------------------------
__global__ void scatter_kernel(const float* __restrict__ phi,
                               const float* __restrict__ h,
                               const int*   __restrict__ senders,
                               const int*   __restrict__ receivers,
                               float*       __restrict__ A) {
    int gt   = blockIdx.x * blockDim.x + threadIdx.x;
    int e    = gt >> 5;
    int lane = gt & 31;
    if (e >= N_EDGES) return;
    int s = senders[e], r = receivers[e];
    int c    = lane & 15;
    int half = lane >> 4;
    float hc = h[s * CCH + c];
    const float4* pe4 =
        reinterpret_cast<const float4*>(phi + (size_t)e * 64 + half * 32);
    float* Ar = A + (size_t)r * 1024 + (size_t)half * 512 + c;
#pragma unroll
    for (int q = 0; q < 8; ++q) {
        float4 pv = pe4[q];
        float* dst = Ar + q * 64;
        atomic_add_f32(dst +  0, pv.x * hc);
        atomic_add_f32(dst + 16, pv.y * hc);
        atomic_add_f32(dst + 32, pv.z * hc);
        atomic_add_f32(dst + 48, pv.w * hc);
    }
}

// ---------------------------------------------------------------------------
// Invariant contraction + h_new GEMM via V_WMMA_F32_16X16X4_F32.
// One wave32 per 16-atom tile:
//   inv[a, l*64+n*16+c] = (1/sqrt(2l+1)) * sum_m A[a, (lm)*4+n, c]^2
//   h_new[16x16] = (inv_tile[16x256] @ W[256x16]) * cemb     (64 WMMA steps)
// ---------------------------------------------------------------------------
__global__ void inv_gemm_kernel(const float* __restrict__ A,
                                const float* __restrict__ W,      // [256][16]
                                const float* __restrict__ cemb,
                                float*       __restrict__ inv_out,// [N][256]
                                float*       __restrict__ h_out)  // [N][16]
{
    __shared__ float invLDS[16 * 260];   // row pad 260 -> no bank conflicts
    __shared__ float WLDS[KTOT * CCH];
    int abase = blockIdx.x * 16;
    int tid = threadIdx.x;               // 0..31, single wave

    for (int i = tid; i < KTOT * CCH; i += 32) WLDS[i] = W[i];

    const int   lmbase[4] = {0, 1, 4, 9};
    const float rs[4] = {1.0f, 0.57735027f, 0.44721360f, 0.37796447f};
    for (int idx = tid; idx < 16 * 256; idx += 32) {
        int al = idx >> 8;
        int kt = idx & 255;
        int l = kt >> 6, n = (kt >> 4) & 3, c = kt & 15;
        const float* Aa = A + (size_t)(abase + al) * 1024;
        float acc = 0.0f;
        int mc = 2 * l + 1, base = lmbase[l];
        for (int m = 0; m < mc; ++m) {
            float v = Aa[((base + m) * 4 + n) * 16 + c];
            acc += v * v;
        }
        acc *= rs[l];
        invLDS[al * 260 + kt] = acc;
        inv_out[(size_t)(abase + al) * KTOT + kt] = acc;
    }
    __syncthreads();

    // WMMA f32 16x16x4, 64 k-steps. ISA 7.12.2 layout:
    //   A (16x4): lanes 0-15 rows M, v[0]=K0,v[1]=K1; lanes 16-31 v[0]=K2,v[1]=K3
    //   B (4x16): lanes 0-15 cols N of rows K0/K1; lanes 16-31 rows K2/K3
    int row = tid & 15;
    int hi  = tid >> 4;                  // 0 or 1
    v8f acc = {};
#pragma unroll 4
    for (int kk = 0; kk < 64; ++kk) {
        int kb = kk * 4 + 2 * hi;
        v2f a, b;
        a.x = invLDS[row * 260 + kb + 0];
        a.y = invLDS[row * 260 + kb + 1];
        b.x = WLDS[(kb + 0) * 16 + row];
        b.y = WLDS[(kb + 1) * 16 + row];
        acc = __builtin_amdgcn_wmma_f32_16x16x4_f32(
                  false, a, false, b, (short)0, acc, false, false);
    }
    // D layout: VGPR v -> row M = v + 8*hi, col N = row
#pragma unroll
    for (int v = 0; v < 8; ++v) {
        int aidx = abase + v + hi * 8;
        h_out[aidx * CCH + row] = acc[v] * cemb[aidx * CCH + row];
    }
}

// ---------------------------------------------------------------------------
// e_atom = inv2 @ w_out + comp_weights[species]; segment-sum into out[8].
// One wave32 per atom: lane-strided coalesced loads + shfl_xor reduction.
// ---------------------------------------------------------------------------
__global__ void final_kernel(const float* __restrict__ inv2,
                             const float* __restrict__ w_out,
                             const float* __restrict__ comp_w,
                             const int*   __restrict__ species,
                             const int*   __restrict__ sids,
                             float*       __restrict__ out) {
    int gt   = blockIdx.x * blockDim.x + threadIdx.x;
    int a    = gt >> 5;
    int lane = gt & 31;
    if (a >= N_ATOMS) return;
    const float* ia = inv2 + (size_t)a * KTOT;
    float acc = 0.0f;
#pragma unroll
    for (int i = 0; i < KTOT / 32; ++i) {
        int k = i * 32 + lane;
        acc += ia[k] * w_out[k];
    }
#pragma unroll
    for (int off = 16; off > 0; off >>= 1)
        acc += __shfl_xor(acc, off, 32);
    if (lane == 0) {
        acc += comp_w[species[a]];
        atomic_add_f32(&out[sids[a]], acc);
    }
}

// ---------------------------------------------------------------------------
// Host launch
// ---------------------------------------------------------------------------
extern "C" void kernel_launch(void* const* d_in, const int* in_sizes, int n_in,
                              void* d_out, int out_size, void* d_ws, size_t ws_size,
                              hipStream_t stream) {
    const float* positions = (const float*)d_in[0];
    const float* embed     = (const float*)d_in[1];
    const float* W_rad     = (const float*)d_in[2];
    const float* W_inv1    = (const float*)d_in[3];
    const float* W_inv2    = (const float*)d_in[4];
    const float* w_out     = (const float*)d_in[5];
    const float* comp_w    = (const float*)d_in[6];
    const int*   senders   = (const int*)d_in[7];
    const int*   receivers = (const int*)d_in[8];
    const int*   species   = (const int*)d_in[9];
    const int*   sids      = (const int*)d_in[10];
    float* out = (float*)d_out;

    float* ws   = (float*)d_ws;
    float* phi  = ws;                                  // E*64   = 10.24M f
    float* A    = phi  + (size_t)N_EDGES * 64;         // N*1024 = 10.24M f
    float* cemb = A    + (size_t)N_ATOMS * 1024;       // N*16
    float* h1   = cemb + (size_t)N_ATOMS * CCH;        // N*16
    float* invb = h1   + (size_t)N_ATOMS * CCH;        // N*256  =  2.56M f
    float* h2   = invb + (size_t)N_ATOMS * KTOT;       // N*16

    const size_t nA4 = (size_t)N_ATOMS * 1024 / 4;

    // edge geometry (shared by both layers)
    edge_phi_kernel<<<(N_EDGES + 127) / 128, 128, 0, stream>>>(
        positions, W_rad, senders, receivers, phi);
    // initial node features
    cemb_kernel<<<(N_ATOMS * CCH + 255) / 256, 256, 0, stream>>>(
        embed, species, cemb);

    // ---- layer 1 ----
    zero4_kernel<<<1024, 256, 0, stream>>>((float4*)A, nA4);
    scatter_kernel<<<N_EDGES * 32 / 256, 256, 0, stream>>>(
        phi, cemb, senders, receivers, A);
    inv_gemm_kernel<<<N_ATOMS / 16, 32, 0, stream>>>(A, W_inv1, cemb, invb, h1);

    // ---- layer 2 ----
    zero4_kernel<<<1024, 256, 0, stream>>>((float4*)A, nA4);
    scatter_kernel<<<N_EDGES * 32 / 256, 256, 0, stream>>>(
        phi, h1, senders, receivers, A);
    inv_gemm_kernel<<<N_ATOMS / 16, 32, 0, stream>>>(A, W_inv2, cemb, invb, h2);

    // ---- readout ----
    zero_kernel<<<1, 32, 0, stream>>>(out, N_STRUCT);
    final_kernel<<<(N_ATOMS * 32 + 255) / 256, 256, 0, stream>>>(
        invb, w_out, comp_w, species, sids, out);
}